// GNN_foraging_simple_71536975282839
// MI455X (gfx1250) — compile-verified
//
#include <hip/hip_runtime.h>
#include <hip/hip_bf16.h>

typedef __attribute__((ext_vector_type(2))) float v2f;
typedef __attribute__((ext_vector_type(8))) float v8f;

// ---------------------------------------------------------------------------
// Native f32 global atomic add (global_atomic_add_f32, no CAS loop)
// ---------------------------------------------------------------------------
__device__ __forceinline__ void atomAddF32(float* p, float v) {
    unsafeAtomicAdd(p, v);
}

// ---------------------------------------------------------------------------
// Degree / normalization kernels
// ---------------------------------------------------------------------------
__global__ void k_deg_init(float* __restrict__ deg, int n) {
    int i = blockIdx.x * blockDim.x + threadIdx.x;
    if (i < n) deg[i] = 1.0f;  // +1 self loop
}

__global__ void k_deg_edges(float* __restrict__ deg, const int* __restrict__ dst, int ne) {
    int i = blockIdx.x * blockDim.x + threadIdx.x;
    if (i < ne) atomAddF32(&deg[dst[i]], 1.0f);
}

__global__ void k_deg_rsqrt(float* __restrict__ deg, int n) {
    int i = blockIdx.x * blockDim.x + threadIdx.x;
    if (i < n) deg[i] = rsqrtf(deg[i]);
}

// ---------------------------------------------------------------------------
// Zero a float buffer (float4 grid-stride)
// ---------------------------------------------------------------------------
__global__ void k_zero(float* __restrict__ p, size_t n4) {
    size_t i = (size_t)blockIdx.x * blockDim.x + threadIdx.x;
    size_t stride = (size_t)gridDim.x * blockDim.x;
    float4 z = make_float4(0.f, 0.f, 0.f, 0.f);
    float4* p4 = (float4*)p;
    for (; i < n4; i += stride) p4[i] = z;
}

// ---------------------------------------------------------------------------
// WMMA fp32 GEMM:  out[M x NCOLS] = A[M x 128] * W[128 x NCOLS] (+ bias)
// Block covers a 32-row M tile (two 16-row WMMA tiles -> two independent
// accumulator chains per wave to hide XDL latency; B fragment shared).
// Each wave owns one 16-col N tile: blockDim.x = (NCOLS/16)*32.
// NCOLS is a compile-time constant so all B loads / C stores use immediate
// offsets off a single per-lane base pointer (no 64-bit address math in loop).
// 100000 % 32 == 0 -> no M bounds checks.
// ---------------------------------------------------------------------------
#define GEMM_K   128
#define GEMM_LDK 132   // padded LDS stride: 132 % 64 == 4 -> conflict-free

template<int NCOLS>
__global__ void k_gemm_wmma(const float* __restrict__ A,
                            const float* __restrict__ W,
                            const float* __restrict__ bias,
                            float* __restrict__ out) {
    constexpr int ROWS = 32;
    __shared__ __align__(16) float ldsA[ROWS * GEMM_LDK];

    const int mbase = blockIdx.x * ROWS;

    // Cooperative float4 load of the 32x128 A tile into padded LDS.
    for (int i = threadIdx.x; i < ROWS * (GEMM_K / 4); i += blockDim.x) {
        const int r = i >> 5;          // row (32 float4 per row)
        const int q = i & 31;          // float4 index within row
        const float4 v = ((const float4*)(A + (mbase + r) * GEMM_K))[q];
        *(float4*)(&ldsA[r * GEMM_LDK + q * 4]) = v;
    }
    __syncthreads();

    const int wave  = threadIdx.x >> 5;
    const int lane  = threadIdx.x & 31;
    const int half  = lane >> 4;   // 0: K pair (0,1); 1: K pair (2,3)
    const int idx   = lane & 15;   // M row within tile for A, N col for B
    const int ntile = wave * 16;

    const float* Wp  = W + 2 * half * NCOLS + ntile + idx;       // per-lane B base
    const float* Ap0 = &ldsA[idx * GEMM_LDK + 2 * half];         // tile 0 rows
    const float* Ap1 = &ldsA[(idx + 16) * GEMM_LDK + 2 * half];  // tile 1 rows

    v8f c0 = {}, c1 = {};
#pragma unroll
    for (int kk = 0; kk < GEMM_K; kk += 4) {
        v2f a0, a1, b;
        a0.x = Ap0[kk];
        a0.y = Ap0[kk + 1];
        a1.x = Ap1[kk];
        a1.y = Ap1[kk + 1];
        b.x  = Wp[kk * NCOLS];          // constant immediate offsets
        b.y  = Wp[(kk + 1) * NCOLS];
        c0 = __builtin_amdgcn_wmma_f32_16x16x4_f32(false, a0, false, b,
                                                   (short)0, c0, false, false);
        c1 = __builtin_amdgcn_wmma_f32_16x16x4_f32(false, a1, false, b,
                                                   (short)0, c1, false, false);
    }

    const float bval = bias ? bias[ntile + idx] : 0.0f;
    float* outp = out + mbase * NCOLS + ntile + idx;
#pragma unroll
    for (int v = 0; v < 8; ++v) {
        const int m = v + 8 * half;    // C layout: VGPR v = row v (+8 upper half)
        outp[m * NCOLS]        = c0[v] + bval;
        outp[(m + 16) * NCOLS] = c1[v] + bval;
    }
}

// ---------------------------------------------------------------------------
// Edge scatter: agg[dst] += h[src] * dinv[src]*dinv[dst]
// One wave per edge; each lane handles 4 consecutive features (float4 load,
// 4 native f32 atomics). 8 waves (edges) per 256-thread block.
// ---------------------------------------------------------------------------
__global__ void k_scatter(const float* __restrict__ h,
                          const int* __restrict__ src,
                          const int* __restrict__ dst,
                          const float* __restrict__ dinv,
                          float* __restrict__ agg, int ne) {
    const int e = blockIdx.x * (blockDim.x >> 5) + (threadIdx.x >> 5);
    if (e >= ne) return;
    const int lane = threadIdx.x & 31;
    const int s = src[e];
    const int d = dst[e];
    const float norm = dinv[s] * dinv[d];
    const float4 v = *(const float4*)(h + s * 128 + lane * 4);
    float* base = agg + d * 128 + lane * 4;
    atomAddF32(base + 0, v.x * norm);
    atomAddF32(base + 1, v.y * norm);
    atomAddF32(base + 2, v.z * norm);
    atomAddF32(base + 3, v.w * norm);
}

// ---------------------------------------------------------------------------
// Epilogue: agg = relu(agg + h * dinv^2 + bias)    (feature dim = 128)
// ---------------------------------------------------------------------------
__global__ void k_self_bias_relu(float* __restrict__ agg,
                                 const float* __restrict__ h,
                                 const float* __restrict__ dinv,
                                 const float* __restrict__ bias, int n_nodes) {
    const size_t total = (size_t)n_nodes * 128;
    const size_t stride = (size_t)gridDim.x * blockDim.x;
    for (size_t i = (size_t)blockIdx.x * blockDim.x + threadIdx.x; i < total; i += stride) {
        const int f    = (int)(i & 127);
        const int node = (int)(i >> 7);
        const float di = dinv[node];
        float v = agg[i] + h[i] * (di * di) + bias[f];
        agg[i] = v > 0.f ? v : 0.f;
    }
}

// ---------------------------------------------------------------------------
// Host-side pipeline
// ---------------------------------------------------------------------------
extern "C" void kernel_launch(void* const* d_in, const int* in_sizes, int n_in,
                              void* d_out, int out_size, void* d_ws, size_t ws_size,
                              hipStream_t stream) {
    const float* x   = (const float*)d_in[0];
    const int*   ei  = (const int*)d_in[1];
    const float* W1  = (const float*)d_in[2];
    const float* b1  = (const float*)d_in[3];
    const float* W2  = (const float*)d_in[4];
    const float* b2  = (const float*)d_in[5];
    const float* Wfc = (const float*)d_in[6];
    const float* bfc = (const float*)d_in[7];
    float* out = (float*)d_out;

    const int n_nodes = in_sizes[0] / 128;  // 100000
    const int n_edges = in_sizes[1] / 2;    // 1600000
    const int hid     = in_sizes[3];        // 128
    const int* src = ei;
    const int* dst = ei + n_edges;

    float* h0   = (float*)d_ws;                      // [N,128] linear output
    float* h1   = h0 + (size_t)n_nodes * hid;        // [N,128] aggregation / layer out
    float* dinv = h1 + (size_t)n_nodes * hid;        // [N] degree -> dinv

    const int mtiles = n_nodes / 32;                 // 3125 (exact)
    const int ewaves_blocks = (n_edges + 7) / 8;     // 8 edges/block
    const size_t feat4 = (size_t)n_nodes * hid / 4;

    // --- degree normalization ---
    k_deg_init <<<(n_nodes + 255) / 256, 256, 0, stream>>>(dinv, n_nodes);
    k_deg_edges<<<(n_edges + 255) / 256, 256, 0, stream>>>(dinv, dst, n_edges);
    k_deg_rsqrt<<<(n_nodes + 255) / 256, 256, 0, stream>>>(dinv, n_nodes);

    // --- layer 1: h0 = X @ W1 ; h1 = relu(norm-agg(h0) + self + b1) ---
    k_gemm_wmma<128><<<mtiles, 256, 0, stream>>>(x, W1, nullptr, h0);
    k_zero<<<1024, 256, 0, stream>>>(h1, feat4);
    k_scatter<<<ewaves_blocks, 256, 0, stream>>>(h0, src, dst, dinv, h1, n_edges);
    k_self_bias_relu<<<4096, 256, 0, stream>>>(h1, h0, dinv, b1, n_nodes);

    // --- layer 2: h0 = h1 @ W2 ; h1 = relu(norm-agg(h0) + self + b2) ---
    k_gemm_wmma<128><<<mtiles, 256, 0, stream>>>(h1, W2, nullptr, h0);
    k_zero<<<1024, 256, 0, stream>>>(h1, feat4);
    k_scatter<<<ewaves_blocks, 256, 0, stream>>>(h0, src, dst, dinv, h1, n_edges);
    k_self_bias_relu<<<4096, 256, 0, stream>>>(h1, h0, dinv, b2, n_nodes);

    // --- final FC: out = h1 @ Wfc + bfc ---
    k_gemm_wmma<64><<<mtiles, 128, 0, stream>>>(h1, Wfc, bfc, out);
}